// ATTENTION_CNN_70806830841953
// MI455X (gfx1250) — compile-verified
//
#include <hip/hip_runtime.h>
#include <hip/hip_bf16.h>
#include <cstddef>

// ---------------------------------------------------------------------------
// MI455X / gfx1250 implementation.
//  - Flash-attention (online softmax) with V_WMMA_F32_16X16X4_F32 (fp32 WMMA,
//    K=4 == head dim granularity) so no N x N energy matrix is materialized.
//  - All intermediates (~18 MB) fit in the 192 MB L2.
//  - wave32: lane = tid & 31, xor-16 shuffles couple the two 16-lane halves
//    of the WMMA C/D layout.
// ---------------------------------------------------------------------------

typedef __attribute__((ext_vector_type(2))) float v2f;
typedef __attribute__((ext_vector_type(8))) float v8f;

__device__ __forceinline__ v8f wmma_f32_4(v2f a, v2f b, v8f c) {
  // D(16x16) = A(16x4) * B(4x16) + C ; fp32 in/out.
  return __builtin_amdgcn_wmma_f32_16x16x4_f32(
      /*neg_a=*/false, a, /*neg_b=*/false, b,
      /*c_mod=*/(short)0, c, /*reuse_a=*/false, /*reuse_b=*/false);
}

// ---------------------------------------------------------------------------
// 3x3 VALID conv, stride 1.  grid = (cdiv(Hout*Wout,256), Cout), block 256.
// Weights for this block's output channel staged in LDS.
// ---------------------------------------------------------------------------
__global__ void conv3x3_bias(const float* __restrict__ in,
                             const float* __restrict__ w,
                             const float* __restrict__ bias,
                             float* __restrict__ out,
                             int Cin, int Hin, int Win, int Hout, int Wout) {
  extern __shared__ float sw[];  // Cin * 9
  const int oc = blockIdx.y;
  const int nw = Cin * 9;
  for (int i = threadIdx.x; i < nw; i += blockDim.x) sw[i] = w[oc * nw + i];
  __syncthreads();

  const int HW = Hout * Wout;
  const int idx = blockIdx.x * blockDim.x + threadIdx.x;
  if (idx >= HW) return;
  const int y = idx / Wout;
  const int x = idx - y * Wout;

  float acc = bias[oc];
  for (int ci = 0; ci < Cin; ++ci) {
    const float* ip = in + ((size_t)ci * Hin + y) * Win + x;
    const float* wp = sw + ci * 9;
#pragma unroll
    for (int ky = 0; ky < 3; ++ky)
#pragma unroll
      for (int kx = 0; kx < 3; ++kx)
        acc = fmaf(ip[ky * Win + kx], wp[ky * 3 + kx], acc);
  }
  out[(size_t)oc * HW + idx] = acc;
}

// ---------------------------------------------------------------------------
// Per-channel batch stats (biased variance).  grid = C blocks, block 256.
// meanvar[2c] = mean, meanvar[2c+1] = var.
// ---------------------------------------------------------------------------
__global__ void bn_stats(const float* __restrict__ x,
                         float* __restrict__ meanvar, int HW) {
  __shared__ float ss[256];
  __shared__ float sq[256];
  const int c = blockIdx.x;
  const float* p = x + (size_t)c * HW;
  float s = 0.0f, q = 0.0f;
  for (int i = threadIdx.x; i < HW; i += blockDim.x) {
    float v = p[i];
    s += v;
    q += v * v;
  }
  ss[threadIdx.x] = s;
  sq[threadIdx.x] = q;
  __syncthreads();
  for (int st = 128; st > 0; st >>= 1) {
    if (threadIdx.x < st) {
      ss[threadIdx.x] += ss[threadIdx.x + st];
      sq[threadIdx.x] += sq[threadIdx.x + st];
    }
    __syncthreads();
  }
  if (threadIdx.x == 0) {
    float m = ss[0] / (float)HW;
    meanvar[2 * c] = m;
    meanvar[2 * c + 1] = sq[0] / (float)HW - m * m;
  }
}

// ---------------------------------------------------------------------------
// Fused BN (train-mode) + ReLU + 2x2 maxpool.  grid = (cdiv(Hp*Wp,256), C).
// ---------------------------------------------------------------------------
__global__ void bn_relu_pool(const float* __restrict__ x,
                             const float* __restrict__ mv,
                             const float* __restrict__ g,
                             const float* __restrict__ b,
                             float* __restrict__ out,
                             int Hin, int Win, int Hp, int Wp) {
  const int c = blockIdx.y;
  const int HWp = Hp * Wp;
  const int idx = blockIdx.x * blockDim.x + threadIdx.x;
  if (idx >= HWp) return;
  const int py = idx / Wp;
  const int px = idx - py * Wp;

  const float mean = mv[2 * c];
  const float var = mv[2 * c + 1];
  const float inv = rsqrtf(var + 1e-5f);
  const float scale = g[c] * inv;
  const float shift = b[c] - mean * scale;

  const float* p = x + ((size_t)c * Hin + 2 * py) * Win + 2 * px;
  float m = -1e30f;
#pragma unroll
  for (int dy = 0; dy < 2; ++dy)
#pragma unroll
    for (int dx = 0; dx < 2; ++dx) {
      float vv = fmaxf(fmaf(scale, p[dy * Win + dx], shift), 0.0f);
      m = fmaxf(m, vv);
    }
  out[(size_t)c * HWp + idx] = m;
}

// ---------------------------------------------------------------------------
// 1x1 q/k/v projections.  x: [C,N] -> q,k: [D,NP], v: [C,NP] (NP = padded
// stride, zero-filled in [N,NP) so flash tail tiles read zeros).
// grid = cdiv(NP, 256), block 256.
// ---------------------------------------------------------------------------
template <int C, int D>
__global__ void qkv_proj(const float* __restrict__ x,
                         const float* __restrict__ qw, const float* __restrict__ qb,
                         const float* __restrict__ kw, const float* __restrict__ kb,
                         const float* __restrict__ vw, const float* __restrict__ vb,
                         float* __restrict__ q, float* __restrict__ k,
                         float* __restrict__ v, int N, int NP) {
  const int n = blockIdx.x * blockDim.x + threadIdx.x;
  if (n >= NP) return;
  if (n >= N) {  // zero padding region
#pragma unroll
    for (int d = 0; d < D; ++d) {
      q[(size_t)d * NP + n] = 0.0f;
      k[(size_t)d * NP + n] = 0.0f;
    }
    for (int o = 0; o < C; ++o) v[(size_t)o * NP + n] = 0.0f;
    return;
  }
  float qa[D], ka[D], va[C];
#pragma unroll
  for (int d = 0; d < D; ++d) {
    qa[d] = qb[d];
    ka[d] = kb[d];
  }
  for (int o = 0; o < C; ++o) va[o] = vb[o];
  for (int ci = 0; ci < C; ++ci) {
    const float xv = x[(size_t)ci * N + n];
#pragma unroll
    for (int d = 0; d < D; ++d) {
      qa[d] = fmaf(qw[d * C + ci], xv, qa[d]);
      ka[d] = fmaf(kw[d * C + ci], xv, ka[d]);
    }
    for (int o = 0; o < C; ++o) va[o] = fmaf(vw[o * C + ci], xv, va[o]);
  }
#pragma unroll
  for (int d = 0; d < D; ++d) {
    q[(size_t)d * NP + n] = qa[d];
    k[(size_t)d * NP + n] = ka[d];
  }
  for (int o = 0; o < C; ++o) v[(size_t)o * NP + n] = va[o];
}

// ---------------------------------------------------------------------------
// Flash attention, fp32 WMMA.  One wave32 per 16-query tile.
//   Et[m,n]  = WMMA(A = K-tile [16m x 4c], B = Q-tile [4c x 16n])  (xDSLICES)
//   softmax rows (over m) = vgpr-index dim + one xor-16 shuffle
//   Ot[c,n] += WMMA(A = V-tile [16c x 4m], B = Pt-slice [4m x 16n]) (x4xCHALF)
// Epilogue: out = gamma * Ot / l + xin   (residual), stride N (contiguous).
// DSLICES = headdim/4, CHALF = C/16.  block = 128 (4 waves).
// ---------------------------------------------------------------------------
template <int DSLICES, int CHALF>
__global__ __launch_bounds__(128) void flash_attn(
    const float* __restrict__ q, const float* __restrict__ k,
    const float* __restrict__ v, const float* __restrict__ xin,
    float* __restrict__ out, const float* __restrict__ gamma_p,
    int N, int NP, int ntiles) {
  const int tid = threadIdx.x;
  const int wave = tid >> 5;
  const int lane = tid & 31;
  const int tile = blockIdx.x * 4 + wave;
  if (tile >= ntiles) return;  // wave-uniform: EXEC stays all-ones for WMMA

  const bool hi = lane >= 16;
  const int lan16 = lane & 15;
  const int n0 = tile * 16;
  const int nIdx = n0 + lan16;
  const float gamma = gamma_p[0];

  // Stationary Q operand (B-matrix layout, mirrors the 16x4 A layout).
  v2f qB[DSLICES];
#pragma unroll
  for (int j = 0; j < DSLICES; ++j) {
    const int c0 = 4 * j + (hi ? 2 : 0);
    const int c1 = 4 * j + (hi ? 3 : 1);
    qB[j].x = q[(size_t)c0 * NP + nIdx];
    qB[j].y = q[(size_t)c1 * NP + nIdx];
  }

  v8f acc[CHALF];
  const v8f vzero = {};
#pragma unroll
  for (int ch = 0; ch < CHALF; ++ch) acc[ch] = vzero;
  float m_run = -1e30f;
  float l_run = 0.0f;

  for (int m0 = 0; m0 < NP; m0 += 16) {
    // --- energy^T tile via fp32 WMMA --------------------------------------
    v8f e = vzero;
#pragma unroll
    for (int j = 0; j < DSLICES; ++j) {
      const int c0 = 4 * j + (hi ? 2 : 0);
      const int c1 = 4 * j + (hi ? 3 : 1);
      const int m = m0 + lan16;
      v2f kA;
      kA.x = k[(size_t)c0 * NP + m];
      kA.y = k[(size_t)c1 * NP + m];
      e = wmma_f32_4(kA, qB[j], e);
    }

    // --- online softmax (rows = m = vgpr index dim + other half) ----------
    float p[8];
    const int mbase = m0 + (hi ? 8 : 0);
    float lmax = -1e30f;
#pragma unroll
    for (int r = 0; r < 8; ++r) {
      float ev = (mbase + r < N) ? e[r] : -1e30f;  // mask tail keys
      p[r] = ev;
      lmax = fmaxf(lmax, ev);
    }
    const float tmax = fmaxf(lmax, __shfl_xor(lmax, 16, 32));
    const float m_new = fmaxf(m_run, tmax);
    const float scale = __expf(m_run - m_new);
    float lsum = 0.0f;
#pragma unroll
    for (int r = 0; r < 8; ++r) {
      p[r] = __expf(p[r] - m_new);
      lsum += p[r];
    }
    l_run = l_run * scale + (lsum + __shfl_xor(lsum, 16, 32));
    m_run = m_new;
#pragma unroll
    for (int ch = 0; ch < CHALF; ++ch)
#pragma unroll
      for (int r = 0; r < 8; ++r) acc[ch][r] *= scale;

    // --- C/D layout -> B-operand layout for P^T (4 K-slices of 4 rows) ----
    v2f bP[4];
#pragma unroll
    for (int jj = 0; jj < 2; ++jj) {  // rows 0..7: lo lanes own them
      const float sx = __shfl_xor(p[4 * jj + 2], 16, 32);
      const float sy = __shfl_xor(p[4 * jj + 3], 16, 32);
      bP[jj].x = hi ? sx : p[4 * jj + 0];
      bP[jj].y = hi ? sy : p[4 * jj + 1];
    }
#pragma unroll
    for (int jj = 0; jj < 2; ++jj) {  // rows 8..15: hi lanes own them
      const float sx = __shfl_xor(p[4 * jj + 0], 16, 32);
      const float sy = __shfl_xor(p[4 * jj + 1], 16, 32);
      bP[2 + jj].x = hi ? p[4 * jj + 2] : sx;
      bP[2 + jj].y = hi ? p[4 * jj + 3] : sy;
    }

    // --- O^T += V_tile x P^T  (4 chained K=4 WMMAs per 16-channel half) ---
#pragma unroll
    for (int ch = 0; ch < CHALF; ++ch) {
      const int crow = ch * 16 + lan16;
#pragma unroll
      for (int j = 0; j < 4; ++j) {
        const int mA = m0 + 4 * j + (hi ? 2 : 0);  // even, NP mult of 16 -> 8B aligned
        const v2f aV = *(const v2f*)(v + (size_t)crow * NP + mA);
        acc[ch] = wmma_f32_4(aV, bP[j], acc[ch]);
      }
    }
  }

  // --- epilogue: normalize, gamma * out + residual ------------------------
  const float rl = 1.0f / l_run;
  if (nIdx < N) {
#pragma unroll
    for (int ch = 0; ch < CHALF; ++ch)
#pragma unroll
      for (int r = 0; r < 8; ++r) {
        const int c = ch * 16 + r + (hi ? 8 : 0);
        const size_t o = (size_t)c * N + nIdx;
        out[o] = fmaf(gamma, acc[ch][r] * rl, xin[o]);
      }
  }
}

// ---------------------------------------------------------------------------
// FC: out[o] = fc_w[o,:] . h + fc_b[o].  grid = 50 blocks, block 256.
// ---------------------------------------------------------------------------
__global__ void fc_kernel(const float* __restrict__ h,
                          const float* __restrict__ w,
                          const float* __restrict__ b,
                          float* __restrict__ out, int K) {
  __shared__ float red[256];
  const int o = blockIdx.x;
  const float* wr = w + (size_t)o * K;
  float s = 0.0f;
  for (int i = threadIdx.x; i < K; i += blockDim.x) s = fmaf(wr[i], h[i], s);
  red[threadIdx.x] = s;
  __syncthreads();
  for (int st = 128; st > 0; st >>= 1) {
    if (threadIdx.x < st) red[threadIdx.x] += red[threadIdx.x + st];
    __syncthreads();
  }
  if (threadIdx.x == 0) out[o] = red[0] + b[o];
}

// ---------------------------------------------------------------------------
// Host-side orchestration (all on `stream`, graph-capture safe).
// ---------------------------------------------------------------------------
static inline int cdiv(int a, int b) { return (a + b - 1) / b; }

extern "C" void kernel_launch(void* const* d_in, const int* in_sizes, int n_in,
                              void* d_out, int out_size, void* d_ws, size_t ws_size,
                              hipStream_t stream) {
  (void)in_sizes; (void)n_in; (void)out_size; (void)ws_size;

  const float* x       = (const float*)d_in[0];   // [1,3,256,256]
  const float* conv1_w = (const float*)d_in[1];   // [32,3,3,3]
  const float* conv1_b = (const float*)d_in[2];
  const float* bn1_g   = (const float*)d_in[3];
  const float* bn1_b   = (const float*)d_in[4];
  const float* a1_qw   = (const float*)d_in[5];   // [4,32]
  const float* a1_qb   = (const float*)d_in[6];
  const float* a1_kw   = (const float*)d_in[7];
  const float* a1_kb   = (const float*)d_in[8];
  const float* a1_vw   = (const float*)d_in[9];   // [32,32]
  const float* a1_vb   = (const float*)d_in[10];
  const float* a1_gamma= (const float*)d_in[11];
  const float* conv2_w = (const float*)d_in[12];  // [64,32,3,3]
  const float* conv2_b = (const float*)d_in[13];
  const float* bn2_g   = (const float*)d_in[14];
  const float* bn2_b   = (const float*)d_in[15];
  const float* a2_qw   = (const float*)d_in[16];  // [8,64]
  const float* a2_qb   = (const float*)d_in[17];
  const float* a2_kw   = (const float*)d_in[18];
  const float* a2_kb   = (const float*)d_in[19];
  const float* a2_vw   = (const float*)d_in[20];  // [64,64]
  const float* a2_vb   = (const float*)d_in[21];
  const float* a2_gamma= (const float*)d_in[22];
  const float* fc_w    = (const float*)d_in[23];  // [50, 246016]
  const float* fc_b    = (const float*)d_in[24];
  float* out = (float*)d_out;                     // [1,50]

  // Dimensions
  const int N1 = 127 * 127;            // 16129
  const int T1 = cdiv(N1, 16);         // 1009 query tiles
  const int NP1 = T1 * 16;             // 16144 padded stride
  const int N2 = 62 * 62;              // 3844
  const int T2 = cdiv(N2, 16);         // 241
  const int NP2 = T2 * 16;             // 3856

  // Workspace layout (floats).  Total ~18.2 MB (c2 aliases c1's slot);
  // everything stays L2-resident on MI455X (192 MB).
  float* ws = (float*)d_ws;
  size_t o = 0;
  float* c1 = ws + o; o += (size_t)32 * 254 * 254;   // 2,064,512  (reused as c2)
  float* mv1 = ws + o; o += 64;
  float* p1 = ws + o; o += (size_t)32 * N1;          // 516,128
  float* q1 = ws + o; o += (size_t)4 * NP1;
  float* k1 = ws + o; o += (size_t)4 * NP1;
  float* v1 = ws + o; o += (size_t)32 * NP1;
  float* a1 = ws + o; o += (size_t)32 * N1;
  float* mv2 = ws + o; o += 128;
  float* p2 = ws + o; o += (size_t)64 * N2;
  float* q2 = ws + o; o += (size_t)8 * NP2;
  float* k2 = ws + o; o += (size_t)8 * NP2;
  float* v2 = ws + o; o += (size_t)64 * NP2;
  float* a2 = ws + o; o += (size_t)64 * N2;
  float* c2 = c1;  // conv1 output dead by the time conv2 runs

  // ---- Stage 1: conv1 -> BN stats -> BN+ReLU+pool -----------------------
  conv3x3_bias<<<dim3(cdiv(254 * 254, 256), 32), 256, 3 * 9 * 4, stream>>>(
      x, conv1_w, conv1_b, c1, 3, 256, 256, 254, 254);
  bn_stats<<<32, 256, 0, stream>>>(c1, mv1, 254 * 254);
  bn_relu_pool<<<dim3(cdiv(N1, 256), 32), 256, 0, stream>>>(
      c1, mv1, bn1_g, bn1_b, p1, 254, 254, 127, 127);

  // ---- Stage 2: attention 1 (C=32, d=4) ---------------------------------
  qkv_proj<32, 4><<<cdiv(NP1, 256), 256, 0, stream>>>(
      p1, a1_qw, a1_qb, a1_kw, a1_kb, a1_vw, a1_vb, q1, k1, v1, N1, NP1);
  flash_attn<1, 2><<<cdiv(T1, 4), 128, 0, stream>>>(
      q1, k1, v1, p1, a1, a1_gamma, N1, NP1, T1);

  // ---- Stage 3: conv2 -> BN stats -> BN+ReLU+pool -----------------------
  conv3x3_bias<<<dim3(cdiv(125 * 125, 256), 64), 256, 32 * 9 * 4, stream>>>(
      a1, conv2_w, conv2_b, c2, 32, 127, 127, 125, 125);
  bn_stats<<<64, 256, 0, stream>>>(c2, mv2, 125 * 125);
  bn_relu_pool<<<dim3(cdiv(N2, 256), 64), 256, 0, stream>>>(
      c2, mv2, bn2_g, bn2_b, p2, 125, 125, 62, 62);

  // ---- Stage 4: attention 2 (C=64, d=8) ---------------------------------
  qkv_proj<64, 8><<<cdiv(NP2, 256), 256, 0, stream>>>(
      p2, a2_qw, a2_qb, a2_kw, a2_kb, a2_vw, a2_vb, q2, k2, v2, N2, NP2);
  flash_attn<2, 4><<<cdiv(T2, 4), 128, 0, stream>>>(
      q2, k2, v2, p2, a2, a2_gamma, N2, NP2, T2);

  // ---- Stage 5: FC -------------------------------------------------------
  fc_kernel<<<50, 256, 0, stream>>>(a2, fc_w, fc_b, out, 64 * N2);
}